// DR_CML_22084721836468
// MI455X (gfx1250) — compile-verified
//
#include <hip/hip_runtime.h>
#include <hip/hip_bf16.h>

typedef __attribute__((ext_vector_type(2))) float v2f;
typedef __attribute__((ext_vector_type(8))) float v8f;

#define BN   512
#define XDN  512
#define KN   32
#define DCOL 48
// workspace layout (float offsets)
#define WS_D    0            // 512 * 48
#define WS_COEF (BN * DCOL)  // 4 * 512

// ---------------------------------------------------------------------------
// Kernel 1: one wave per sample b. Wave 0 of each block first computes
//   Pw[n] = lm_w . W1cat[:,n],  Pb[n] = lm_b . W1cat[:,n]   (n = net*7 + h)
// via chained V_WMMA_F32_16X16X4_F32 into LDS, then all 8 waves do per-b work.
// ---------------------------------------------------------------------------
__global__ void per_sample(const float* __restrict__ x, const float* __restrict__ y,
                           const float* __restrict__ ps_w, const float* __restrict__ ps_b,
                           const float* __restrict__ lm_w, const float* __restrict__ lm_b,
                           const float* __restrict__ mu_w1, const float* __restrict__ mu_b1,
                           const float* __restrict__ mu_w2, const float* __restrict__ mu_b2,
                           const float* __restrict__ lv_w1, const float* __restrict__ lv_b1,
                           const float* __restrict__ lv_w2, const float* __restrict__ lv_b2,
                           const float* __restrict__ mun_w1, const float* __restrict__ mun_b1,
                           const float* __restrict__ mun_w2, const float* __restrict__ mun_b2,
                           const float* __restrict__ lvn_w1, const float* __restrict__ lvn_b1,
                           const float* __restrict__ lvn_w2, const float* __restrict__ lvn_b2,
                           float* __restrict__ ws)
{
    __shared__ float sPW[32];
    __shared__ float sPB[32];

    const int lane = threadIdx.x & 31;
    const int wave = threadIdx.x >> 5;

    // ---- wave 0: weight projection via WMMA (A rows: lm_w, lm_b; B: W1cat) ----
    if (wave == 0) {
        const int half = lane >> 4;
        const int m    = lane & 15;
        const int n0   = m;                       // tile 0 column (always < 28)
        const int n1   = 16 + m;                  // tile 1 column
        const int nn1  = (n1 < 28) ? n1 : 0;      // clamped for safe loads
        const float msk1 = (n1 < 28) ? 1.f : 0.f;
        const float ma   = (m == 0) ? 1.f : 0.f;  // arithmetic masks: force
        const float mb   = (m == 1) ? 1.f : 0.f;  // unconditional A loads

        const float* s0 = (n0 < 7) ? mu_w1 : (n0 < 14) ? lv_w1 : (n0 < 21) ? mun_w1 : lvn_w1;
        const float* s1 = (nn1 < 7) ? mu_w1 : (nn1 < 14) ? lv_w1 : (nn1 < 21) ? mun_w1 : lvn_w1;
        const int h0 = n0 % 7;
        const int h1 = nn1 % 7;

        v8f c0 = {}; v8f c1 = {};
        for (int k = 0; k < XDN; k += 4) {
            const int ka = k + 2 * half;          // this half's K pair
            const float wa0 = lm_w[ka], wa1 = lm_w[ka + 1];
            const float wb0 = lm_b[ka], wb1 = lm_b[ka + 1];
            v2f A;
            A.x = fmaf(wa0, ma, wb0 * mb);        // row0=lm_w, row1=lm_b, else 0
            A.y = fmaf(wa1, ma, wb1 * mb);
            v2f B0, B1;
            B0.x = s0[ka * 7 + h0];
            B0.y = s0[(ka + 1) * 7 + h0];
            B1.x = s1[ka * 7 + h1] * msk1;
            B1.y = s1[(ka + 1) * 7 + h1] * msk1;
            c0 = __builtin_amdgcn_wmma_f32_16x16x4_f32(false, A, false, B0, (short)0, c0, false, false);
            c1 = __builtin_amdgcn_wmma_f32_16x16x4_f32(false, A, false, B1, (short)0, c1, false, false);
        }
        // D layout: lanes 0-15 hold N=lane; M=0 -> c[0], M=1 -> c[1]
        if (lane < 16) {
            sPW[m]      = c0[0];
            sPB[m]      = c0[1];
            sPW[16 + m] = c1[0];   // columns >= 28 are zero by construction
            sPB[16 + m] = c1[1];
        }
    }
    __syncthreads();

    const int b = blockIdx.x * 8 + wave;
    const float* xrow = x + b * XDN;

    // ---- pass 1: row mean + propensity dot (row kept in registers) ----
    float xr[16];
    float sumx = 0.f, psdot = 0.f;
#pragma unroll
    for (int t = 0; t < 15; ++t) {                // j <= 479 < 511: all in-bounds
        const int j = lane + 32 * t;
        const float v = xrow[j];
        xr[t] = v;
        sumx += v;
        psdot += v * ps_w[j];
    }
    {                                             // tail: j in [480, 511]
        const int j = lane + 480;
        const float v = xrow[j];
        xr[15] = v;
        sumx += v;
        const float ms = (j < 511) ? 1.f : 0.f;
        psdot += v * ps_w[(j < 511) ? j : 510] * ms;
    }
#pragma unroll
    for (int off = 16; off; off >>= 1) {
        sumx  += __shfl_xor(sumx,  off, 32);
        psdot += __shfl_xor(psdot, off, 32);
    }
    const float xbar  = sumx * (1.f / 512.f);
    const float treat = xrow[XDN - 1];
    const float f0 = (treat == 0.f) ? 1.f : 0.f;
    const float f1 = (treat == 1.f) ? 1.f : 0.f;
    const float prop = 1.f / (1.f + expf(-(psdot + ps_b[0])));
    const float w0v = 1.f / (1.f - prop + 1e-4f);
    const float w1v = 1.f / (prop + 1e-4f);

    const float cc = 511.f / 512.f;

    // ---- scalar MLPs for mu / lv at (s=xbar, t=1) and (s=a_last, t=cc) ----
    const float a_last = xbar - xrow[XDN - 2] * (1.f / 512.f);
    float mu0 = mu_b2[0], lvr0 = lv_b2[0], mu1 = mu_b2[0], lvr1 = lv_b2[0];
#pragma unroll
    for (int h = 0; h < 7; ++h) {
        const float pwm = sPW[h],     pbm = sPB[h];
        const float pwl = sPW[7 + h], pbl = sPB[7 + h];
        mu0  += fmaxf(fmaf(xbar,   pwm, pbm      + mu_b1[h]), 0.f) * mu_w2[h];
        lvr0 += fmaxf(fmaf(xbar,   pwl, pbl      + lv_b1[h]), 0.f) * lv_w2[h];
        mu1  += fmaxf(fmaf(a_last, pwm, cc * pbm + mu_b1[h]), 0.f) * mu_w2[h];
        lvr1 += fmaxf(fmaf(a_last, pwl, cc * pbl + lv_b1[h]), 0.f) * lv_w2[h];
    }
    const float lv0 = tanhf(lvr0);
    const float lv1 = tanhf(lvr1);

    // ---- preload mun / lvn nets ----
    float pwm[7], cbm[7], w2m[7], pwl[7], cbl[7], w2l[7];
#pragma unroll
    for (int h = 0; h < 7; ++h) {
        pwm[h] = sPW[14 + h]; cbm[h] = fmaf(cc, sPB[14 + h], mun_b1[h]); w2m[h] = mun_w2[h];
        pwl[h] = sPW[21 + h]; cbl[h] = fmaf(cc, sPB[21 + h], lvn_b1[h]); w2l[h] = lvn_w2[h];
    }
    const float b2m = mun_b2[0], b2l = lvn_b2[0];

    // ---- moments over i = 0..510 of the neg-MLP outputs (branchless tail) ----
    float G = 0.f, M1 = 0.f, M2 = 0.f, L = 0.f;
#pragma unroll
    for (int t = 0; t < 16; ++t) {
        const int j = lane + 32 * t;
        const float ms = (j < XDN - 1) ? 1.f : 0.f;   // only lane 31 @ t=15 masked
        const float a = xbar - xr[t] * (1.f / 512.f);
        float am = b2m, al = b2l;
#pragma unroll
        for (int h = 0; h < 7; ++h) {
            am += fmaxf(fmaf(a, pwm[h], cbm[h]), 0.f) * w2m[h];
            al += fmaxf(fmaf(a, pwl[h], cbl[h]), 0.f) * w2l[h];
        }
        const float lvn = tanhf(al);
        const float g   = 0.5f * expf(-lvn) * ms;
        G += g; M1 += am * g; M2 += am * am * g; L += 0.5f * lvn * ms;
    }
#pragma unroll
    for (int off = 16; off; off >>= 1) {
        G  += __shfl_xor(G,  off, 32);
        M1 += __shfl_xor(M1, off, 32);
        M2 += __shfl_xor(M2, off, 32);
        L  += __shfl_xor(L,  off, 32);
    }

    // ---- D[b,k] = 511*positive + M2 - 2 y M1 + y^2 G + L   (lane = k) ----
    const float yk  = y[b * KN + lane];
    const float mu_s = (lane == 0) ? mu0 : mu1;
    const float lv_s = (lane == 0) ? lv0 : lv1;
    const float gp = 0.5f * expf(-lv_s);
    const float dm = mu_s - yk;
    const float pos = -dm * dm * gp - 0.5f * lv_s;
    const float D = 511.f * pos + M2 - 2.f * yk * M1 + yk * yk * G + L;

    float* Drow = ws + WS_D + b * DCOL;
    Drow[lane] = D;
    if (lane < 16) Drow[32 + lane] = (lane == 0) ? 1.f : 0.f;   // ones col + zero pad
    if (lane == 0) {
        ws[WS_COEF + 0 * BN + b] = f0;
        ws[WS_COEF + 1 * BN + b] = f1;
        ws[WS_COEF + 2 * BN + b] = f0 * w0v;
        ws[WS_COEF + 3 * BN + b] = f1 * w1v;
    }
}

// ---------------------------------------------------------------------------
// Kernel 2: S = coef(4x512) @ D(512x48) via chained WMMA, then epilogue.
// Rows: 0=f0, 1=f1, 2=f0*w0, 3=f1*w1; column 32 of D is ones -> row sums.
// ---------------------------------------------------------------------------
__global__ void reduce_wmma(const float* __restrict__ ws, float* __restrict__ out)
{
    const int lane = threadIdx.x & 31;
    const int half = lane >> 4;
    const int m    = lane & 15;
    const int mc   = m & 3;                         // clamped coef row
    const float msA = (m < 4) ? 1.f : 0.f;
    const float* Dmat = ws + WS_D;
    const float* coef = ws + WS_COEF;

    v8f c0 = {}, c1 = {}, c2 = {};
#pragma unroll 4
    for (int k = 0; k < BN; k += 4) {
        const int ka = k + 2 * half;
        v2f A;
        A.x = coef[mc * BN + ka]     * msA;
        A.y = coef[mc * BN + ka + 1] * msA;
        const float* r0 = Dmat + ka * DCOL;
        const float* r1 = r0 + DCOL;
        v2f B0, B1, B2;
        B0.x = r0[m];      B0.y = r1[m];
        B1.x = r0[16 + m]; B1.y = r1[16 + m];
        B2.x = r0[32 + m]; B2.y = r1[32 + m];
        c0 = __builtin_amdgcn_wmma_f32_16x16x4_f32(false, A, false, B0, (short)0, c0, false, false);
        c1 = __builtin_amdgcn_wmma_f32_16x16x4_f32(false, A, false, B1, (short)0, c1, false, false);
        c2 = __builtin_amdgcn_wmma_f32_16x16x4_f32(false, A, false, B2, (short)0, c2, false, false);
    }

    // n0,n1,T0,T1 live in tile2 column 32 -> lane 0, rows 0..3 -> c2[0..3]
    const float n0 = __shfl(c2[0], 0, 32);
    const float n1 = __shfl(c2[1], 0, 32);
    const float T0 = __shfl(c2[2], 0, 32);
    const float T1 = __shfl(c2[3], 0, 32);
    const float d0 = n0 * 511.f, d1 = n1 * 511.f;

    if (lane < 16) {
        {   // k = lane (tile 0)
            const float cmi0 = c0[0] / d0, cmi1 = c0[1] / d1;
            const float dr = 0.5f * (cmi0 + (c0[2] - cmi0 * 511.f * T0) / d0)
                           + 0.5f * (cmi1 + (c0[3] - cmi1 * 511.f * T1) / d1);
            out[lane]      = fabsf(cmi0 + cmi1) * 0.5f;
            out[32 + lane] = fabsf(dr);
        }
        {   // k = 16 + lane (tile 1)
            const float cmi0 = c1[0] / d0, cmi1 = c1[1] / d1;
            const float dr = 0.5f * (cmi0 + (c1[2] - cmi0 * 511.f * T0) / d0)
                           + 0.5f * (cmi1 + (c1[3] - cmi1 * 511.f * T1) / d1);
            out[16 + lane] = fabsf(cmi0 + cmi1) * 0.5f;
            out[48 + lane] = fabsf(dr);
        }
    }
}

// ---------------------------------------------------------------------------
extern "C" void kernel_launch(void* const* d_in, const int* in_sizes, int n_in,
                              void* d_out, int out_size, void* d_ws, size_t ws_size,
                              hipStream_t stream) {
    (void)in_sizes; (void)n_in; (void)out_size; (void)ws_size;
    const float* x      = (const float*)d_in[0];
    const float* y      = (const float*)d_in[1];
    const float* ps_w   = (const float*)d_in[2];
    const float* ps_b   = (const float*)d_in[3];
    const float* lm_w   = (const float*)d_in[4];
    const float* lm_b   = (const float*)d_in[5];
    const float* mu_w1  = (const float*)d_in[6];
    const float* mu_b1  = (const float*)d_in[7];
    const float* mu_w2  = (const float*)d_in[8];
    const float* mu_b2  = (const float*)d_in[9];
    const float* lv_w1  = (const float*)d_in[10];
    const float* lv_b1  = (const float*)d_in[11];
    const float* lv_w2  = (const float*)d_in[12];
    const float* lv_b2  = (const float*)d_in[13];
    const float* mun_w1 = (const float*)d_in[14];
    const float* mun_b1 = (const float*)d_in[15];
    const float* mun_w2 = (const float*)d_in[16];
    const float* mun_b2 = (const float*)d_in[17];
    const float* lvn_w1 = (const float*)d_in[18];
    const float* lvn_b1 = (const float*)d_in[19];
    const float* lvn_w2 = (const float*)d_in[20];
    const float* lvn_b2 = (const float*)d_in[21];
    float* ws  = (float*)d_ws;
    float* out = (float*)d_out;

    per_sample<<<BN / 8, 256, 0, stream>>>(x, y, ps_w, ps_b, lm_w, lm_b,
                                           mu_w1, mu_b1, mu_w2, mu_b2,
                                           lv_w1, lv_b1, lv_w2, lv_b2,
                                           mun_w1, mun_b1, mun_w2, mun_b2,
                                           lvn_w1, lvn_b1, lvn_w2, lvn_b2, ws);
    reduce_wmma<<<1, 32, 0, stream>>>(ws, out);
}